// RGAT_Stack_68281390072290
// MI455X (gfx1250) — compile-verified
//
#include <hip/hip_runtime.h>
#include <hip/hip_bf16.h>
#include <math.h>

// Problem constants (match reference)
constexpr int B = 512, L = 50, D = 768, V = 10000, R = 40, H = 8;

typedef __attribute__((ext_vector_type(2))) float v2f;
typedef __attribute__((ext_vector_type(8))) float v8f;

// One block = one batch element. 4 waves (128 threads).
// Wave w computes the 16x16 q-tile for l in [16w, 16w+16) via f32 WMMA:
//   A (16x4) = prop rows (gathered from embed), B (4x16) = W[h, rc, :] columns.
// Then: bias + ReLU -> LDS, softmax over l per head, abar[l] = mean_h a[h,l],
// and pooled[d] = sum_l abar[l] * val[b,l,d] (coalesced streaming).
__global__ __launch_bounds__(128)
void pool_kernel(const int* __restrict__ ids,      // (B, L)
                 const float* __restrict__ val,    // (B, L, D)
                 const int* __restrict__ rc_ids,   // (B)
                 const float* __restrict__ embed,  // (V, D)
                 const float* __restrict__ fc2_w,  // (H, R, D)
                 const float* __restrict__ fc2_b,  // (H, R)
                 float* __restrict__ pooled)       // (B, D)
{
    const int b    = blockIdx.x;
    const int tid  = threadIdx.x;
    const int wave = tid >> 5;
    const int lane = tid & 31;
    const int rc   = rc_ids[b];

    __shared__ float q_lds[H][64];  // [h][l], l padded 50 -> 64
    __shared__ float abar[64];

    const float* vbase = val + (size_t)b * L * D;

    // Prefetch stage-2 val rows (one prefetch per 128B line) to overlap with WMMA.
    for (int i = tid; i < L * 24; i += 128) {           // 24 lines per 3072B row
        int l = i / 24, c = i % 24;
        __builtin_prefetch(vbase + (size_t)l * D + (size_t)c * 32, 0, 1);
    }

    // ---- Stage 1: q = A * W via V_WMMA_F32_16X16X4_F32 --------------------
    // A fragment (16x4 f32): lane&15 = row M (= local l), lane>>4 selects K pair.
    const int lrow   = wave * 16 + (lane & 15);
    const int lclamp = (lrow < L) ? lrow : (L - 1);          // clamp: garbage rows unused
    const float* aptr = embed + (size_t)ids[b * L + lclamp] * D + ((lane >> 4) << 1);

    // B fragment (4x16 f32): lane&15 = col N (= head h), lane>>4 selects K pair.
    const int h   = lane & 15;
    const int hcl = (h < H) ? h : (H - 1);                   // clamp: garbage cols unused
    const float* bptr = fc2_w + ((size_t)hcl * R + rc) * D + ((lane >> 4) << 1);

    v8f acc = {0.f, 0.f, 0.f, 0.f, 0.f, 0.f, 0.f, 0.f};
    #pragma unroll 4
    for (int k = 0; k < D; k += 4) {
        v2f av = *(const v2f*)(aptr + k);   // VGPR0 = K (+0/+2), VGPR1 = K (+1/+3)
        v2f bv = *(const v2f*)(bptr + k);
        acc = __builtin_amdgcn_wmma_f32_16x16x4_f32(
            /*neg_a=*/false, av, /*neg_b=*/false, bv,
            /*c_mod=*/(short)0, acc, /*reuse_a=*/false, /*reuse_b=*/false);
    }

    // C layout: lanes 0-15 -> M=r, lanes 16-31 -> M=r+8; N = lane&15 = h.
    const float bias = fc2_b[(size_t)hcl * R + rc];
    if (h < H) {
        #pragma unroll
        for (int r = 0; r < 8; ++r) {
            int l = wave * 16 + ((lane >> 4) << 3) + r;
            float qv = acc[r] + bias;
            q_lds[h][l] = (qv > 0.f) ? qv : 0.f;   // ReLU
        }
    }
    __syncthreads();

    // ---- Softmax over l per head (8 threads; tiny) ------------------------
    if (tid < H) {
        float m = -INFINITY;
        for (int l = 0; l < L; ++l) m = fmaxf(m, q_lds[tid][l]);
        float s = 0.f;
        for (int l = 0; l < L; ++l) s += __expf(q_lds[tid][l] - m);
        float inv = 1.f / s;
        for (int l = 0; l < L; ++l) q_lds[tid][l] = __expf(q_lds[tid][l] - m) * inv;
    }
    __syncthreads();

    // abar[l] = mean over heads (mean-over-h commutes with the l-contraction)
    if (tid < L) {
        float s = 0.f;
        #pragma unroll
        for (int hh = 0; hh < H; ++hh) s += q_lds[hh][tid];
        abar[tid] = s * (1.0f / H);
    }
    __syncthreads();

    // ---- Stage 2: pooled[d] = sum_l abar[l] * val[b,l,d] (coalesced) ------
    float accd[6] = {0.f, 0.f, 0.f, 0.f, 0.f, 0.f};   // D/128 = 6 per thread
    for (int l = 0; l < L; ++l) {
        float w = abar[l];
        const float* vr = vbase + (size_t)l * D;
        #pragma unroll
        for (int k = 0; k < 6; ++k)
            accd[k] = fmaf(w, vr[tid + 128 * k], accd[k]);
    }
    #pragma unroll
    for (int k = 0; k < 6; ++k)
        pooled[(size_t)b * D + tid + 128 * k] = accd[k];
}

// logits[b,c] = sum_k cat[b,k] * lin1_w[rc, k, c] + lin1_b[rc, c], k over 2D
__global__ __launch_bounds__(256)
void logits_kernel(const float* __restrict__ head_out,  // (B, D)
                   const float* __restrict__ tail_out,  // (B, D)
                   const int* __restrict__ rc_ids,
                   const float* __restrict__ lin1_w,    // (R, 2D, 2)
                   const float* __restrict__ lin1_b,    // (R, 2)
                   float* __restrict__ out)             // (B, 2)
{
    const int b = blockIdx.x, tid = threadIdx.x;
    const int rc = rc_ids[b];
    const float* Wb = lin1_w + (size_t)rc * (2 * D) * 2;

    float p0 = 0.f, p1 = 0.f;
    for (int k = tid; k < 2 * D; k += 256) {
        float x = (k < D) ? head_out[(size_t)b * D + k]
                          : tail_out[(size_t)b * D + (k - D)];
        p0 = fmaf(x, Wb[2 * k + 0], p0);
        p1 = fmaf(x, Wb[2 * k + 1], p1);
    }
    __shared__ float r0[256], r1[256];
    r0[tid] = p0; r1[tid] = p1;
    __syncthreads();
    for (int s = 128; s > 0; s >>= 1) {
        if (tid < s) { r0[tid] += r0[tid + s]; r1[tid] += r1[tid + s]; }
        __syncthreads();
    }
    if (tid == 0) {
        out[2 * b + 0] = r0[0] + lin1_b[2 * rc + 0];
        out[2 * b + 1] = r1[0] + lin1_b[2 * rc + 1];
    }
}

// loss = -mean_b log_softmax(logits)[label]
__global__ __launch_bounds__(256)
void loss_kernel(const float* __restrict__ logits,  // (B, 2)
                 const int* __restrict__ labels,    // (B)
                 float* __restrict__ loss_out)
{
    const int tid = threadIdx.x;
    float part = 0.f;
    for (int b = tid; b < B; b += 256) {
        float l0 = logits[2 * b], l1 = logits[2 * b + 1];
        float m = fmaxf(l0, l1);
        float lse = m + logf(expf(l0 - m) + expf(l1 - m));
        part += ((labels[b] == 0) ? l0 : l1) - lse;
    }
    __shared__ float red[256];
    red[tid] = part;
    __syncthreads();
    for (int s = 128; s > 0; s >>= 1) {
        if (tid < s) red[tid] += red[tid + s];
        __syncthreads();
    }
    if (tid == 0) loss_out[0] = -red[0] / (float)B;
}

extern "C" void kernel_launch(void* const* d_in, const int* in_sizes, int n_in,
                              void* d_out, int out_size, void* d_ws, size_t ws_size,
                              hipStream_t stream) {
    const int*   head_ids  = (const int*)d_in[0];
    const int*   tail_ids  = (const int*)d_in[1];
    const float* head_val  = (const float*)d_in[2];
    const float* tail_val  = (const float*)d_in[3];
    const int*   rc_ids    = (const int*)d_in[4];
    const int*   nd_labels = (const int*)d_in[5];
    const float* embed     = (const float*)d_in[6];
    const float* fc2_w     = (const float*)d_in[7];
    const float* fc2_b     = (const float*)d_in[8];
    const float* lin1_w    = (const float*)d_in[9];
    const float* lin1_b    = (const float*)d_in[10];

    float* out      = (float*)d_out;           // [0, 2B) logits, [2B] loss
    float* head_out = (float*)d_ws;            // (B, D)
    float* tail_out = head_out + (size_t)B * D;

    pool_kernel<<<B, 128, 0, stream>>>(head_ids, head_val, rc_ids, embed,
                                       fc2_w, fc2_b, head_out);
    pool_kernel<<<B, 128, 0, stream>>>(tail_ids, tail_val, rc_ids, embed,
                                       fc2_w, fc2_b, tail_out);
    logits_kernel<<<B, 256, 0, stream>>>(head_out, tail_out, rc_ids,
                                         lin1_w, lin1_b, out);
    loss_kernel<<<1, 256, 0, stream>>>(out, nd_labels, out + 2 * B);
}